// DiagonalKernelAverage_25031069401675
// MI455X (gfx1250) — compile-verified
//
#include <hip/hip_runtime.h>

#define TSIZE 512
#define NPLANE 256      // 32 batches * 8 channels
#define TILE_COLS 16
#define NTILES 32       // 512 / 16
#define TPITCH 17       // +1 padding: conflict-free row reads (gcd(17,64)=1)

typedef __attribute__((ext_vector_type(2))) float v2f;
typedef __attribute__((ext_vector_type(8))) float v8f;
typedef unsigned int u32;
typedef __attribute__((ext_vector_type(4))) u32 v4u;
typedef __attribute__((ext_vector_type(8))) int v8i;
typedef __attribute__((ext_vector_type(4))) int v4i;

#if __has_builtin(__builtin_amdgcn_tensor_load_to_lds)
#define HAVE_TDM 1
#else
#define HAVE_TDM 0
#endif

#if HAVE_TDM
// TDM 2D descriptor: tile = 16 elems (dim0, contiguous) x 512 rows (dim1),
// tensor row stride = 512 elems, data_size = 4B. LDS padding: 1 DWORD after
// every 16 DWORDs -> pitch-17 rows, matching tile[r*TPITCH + j].
__device__ __forceinline__ void tdm_load_tile(const float* gsrc, float* ldsdst) {
  const unsigned long long ga = (unsigned long long)(size_t)gsrc;
  v4u g0;
  g0[0] = 1u;                                           // count=1, user desc
  g0[1] = (u32)(size_t)(void*)ldsdst;                   // lds_addr (bytes)
  g0[2] = (u32)ga;                                      // global_addr[31:0]
  g0[3] = (u32)((ga >> 32) & 0x01FFFFFFu) | (2u << 30); // ga[56:32] | type=2
  v8i g1;
  g1[0] = (int)((2u << 16)    // data_size = 4B
              | (1u << 20)    // pad_enable
              | (3u << 22));  // pad_interval: 16 DWORDs (pad_amount=0 -> 1 DW)
  g1[1] = (int)(512u << 16);  // tensor_dim0 = 512 (bits [63:48])
  g1[2] = (int)(512u << 16);  // tensor_dim1 = 512 (bits [95:80])
  g1[3] = (int)(16u << 16);   // tile_dim0 = 16    (bits [127:112])
  g1[4] = 512;                // tile_dim1 = 512, tile_dim2 = 0 (unused)
  g1[5] = 512;                // tensor_dim0_stride = 512 elems
  g1[6] = 0;
  g1[7] = 0;
  const v4i z4 = {0, 0, 0, 0};
  const v8i z8 = {0, 0, 0, 0, 0, 0, 0, 0};
  // 6-arg form (clang-23 / therock-10.0 headers): g0, g1, g2, g3, extra, cpol
  __builtin_amdgcn_tensor_load_to_lds(g0, g1, z4, z4, z8, 0);
}
#endif

// One workgroup per (b, ch) plane. 512 threads = 16 wave32s.
// Pass 1: thread t scans column t (coalesced HBM stream), 2 prefix captures.
// Pass 2: TDM double-buffers 16-col tiles into padded LDS; thread t scans
//         row t, 2 prefix captures; combine into 4 histograms A[corner][m].
// Phase 3: waves 0..3 inclusive-scan one histogram each with
//          V_WMMA_F32_16X16X4_F32, divide by (m+1)^2, scatter to out.
__global__ __launch_bounds__(512, 1)
void corner_window_mean_kernel(const float* __restrict__ x,
                               float* __restrict__ out) {
  const int plane = blockIdx.x;
  const int bB = plane >> 3;
  const int ch = plane & 7;
  const int t = threadIdx.x;

  const float* __restrict__ X = x + (size_t)plane * TSIZE * TSIZE;

  __shared__ float tile0[TSIZE * TPITCH];  // 34 KB each
  __shared__ float tile1[TSIZE * TPITCH];
  __shared__ float A[4][TSIZE];            // 4 corner histograms (8 KB)
  __shared__ float Ys[4][256];             // per-wave scan scratch (4 KB)

#if HAVE_TDM
  if (t < 32) tdm_load_tile(X, tile0);     // prefetch tile 0 during pass 1
#endif

  // Diagonal / anti-diagonal elements (derive the 2nd pair of captures).
  const float d1 = X[(size_t)t * TSIZE + t];            // x[t, t]
  const float d2 = X[(size_t)(511 - t) * TSIZE + t];    // x[511-t, t]
  const float d3 = X[(size_t)t * TSIZE + (511 - t)];    // x[t, 511-t]

  // ---------------- Pass 1: column scans (fully coalesced) ----------------
  {
    const int c = t;
    float run = 0.f, P1 = 0.f, P2 = 0.f;
    const int rP1 = c - 1;       // colprefix(c)
    const int rP2 = 510 - c;     // colprefix(511-c)
#pragma unroll 4
    for (int r = 0; r < TSIZE; ++r) {
      if ((r & 7) == 0 && r + 64 < TSIZE)
        __builtin_prefetch(X + (size_t)(r + 64) * TSIZE + c, 0, 0);
      run += X[(size_t)r * TSIZE + c];
      if (r == rP1) P1 = run;
      if (r == rP2) P2 = run;
    }
    const float Tc = run;
    const float P3 = P2 + d2;    // colprefix(512-c)
    const float P4 = P1 + d1;    // colprefix(c+1)
    A[0][c]       = P1;          // TL: col m=c, rows [0, m)
    A[1][511 - c] = P2;          // TR: col 511-m, rows [0, m)
    A[2][c]       = Tc - P3;     // BL: col m=c, rows [512-m, 512)
    A[3][511 - c] = Tc - P4;     // BR: col 511-m, rows [512-m, 512)
  }

  // ------------- Pass 2: row scans via TDM-fed LDS tiles ------------------
  {
    const int r = t;
    float run = 0.f, Q1 = 0.f, Q3 = 0.f;
    const int jQ1 = r;           // rowprefix(r+1)
    const int jQ3 = 511 - r;     // rowprefix(512-r)
    for (int i = 0; i < NTILES; ++i) {
      const int c0 = i * TILE_COLS;
      float* cur = (i & 1) ? tile1 : tile0;
#if HAVE_TDM
      if (t < 32) __builtin_amdgcn_s_wait_tensorcnt(0);
      __syncthreads();           // tile i resident (and pass-1 A[] published)
      if (t < 32 && i + 1 < NTILES)
        tdm_load_tile(X + (size_t)(c0 + TILE_COLS), (i & 1) ? tile0 : tile1);
#else
      __syncthreads();           // prior readers of `cur` are done
#pragma unroll
      for (int k = 0; k < TILE_COLS; ++k) {
        const int e = t + 512 * k;
        const int rr = e >> 4;
        const int jj = e & 15;
        cur[rr * TPITCH + jj] = X[(size_t)rr * TSIZE + c0 + jj];
      }
      __syncthreads();
#endif
#pragma unroll
      for (int jj = 0; jj < TILE_COLS; ++jj) {
        const int j = c0 + jj;
        run += cur[r * TPITCH + jj];
        if (j == jQ1) Q1 = run;
        if (j == jQ3) Q3 = run;
      }
      __syncthreads();           // done reading `cur` before its refill
    }
    const float Tr = run;
    const float Q2 = Q3 - d3;    // rowprefix(511-r)
    const float Q4 = Q1 - d1;    // rowprefix(r)
    A[0][r]       += Q1;         // TL: row m=r, cols [0, m]
    A[1][r]       += Tr - Q2;    // TR: row m=r, cols [511-m, 512)
    A[2][511 - r] += Q3;         // BL: row 511-m, cols [0, m]
    A[3][511 - r] += Tr - Q4;    // BR: row 511-m, cols [511-m, 512)
  }
  __syncthreads();

  // --------- Phase 3: inclusive scans via V_WMMA_F32_16X16X4_F32 ----------
  const int w = t >> 5;          // wave id (wave32)
  if (w < 4) {
    const int lane = t & 31;
    const int col = lane & 15;         // B/C/D column (N), also A-matrix row
    const int hi  = lane >> 4;         // half-lane select
    const int koff = hi ? 2 : 0;       // K-pair offset for this half
    const float* arr = A[w];
    float* ys = Ys[w];

    float carry = 0.f;                 // running total across 256-blocks
#pragma unroll
    for (int bi = 0; bi < 2; ++bi) {
      const int base = bi * 256;
      // Y = L * X : chunk-local inclusive scans (chunks of 16 as columns).
      v8f acc = {0.f, 0.f, 0.f, 0.f, 0.f, 0.f, 0.f, 0.f};
#pragma unroll
      for (int kk = 0; kk < 4; ++kk) {
        const int row = col;
        const int kc = 4 * kk + koff;
        v2f a, bmat;
        a.x = (kc     <= row) ? 1.f : 0.f;   // L: lower-triangular ones
        a.y = (kc + 1 <= row) ? 1.f : 0.f;
        bmat.x = arr[base + 16 * col + kc];
        bmat.y = arr[base + 16 * col + kc + 1];
        acc = __builtin_amdgcn_wmma_f32_16x16x4_f32(
            false, a, false, bmat, (short)0, acc, false, false);
      }
      // Publish Y so every lane can read the 16 chunk sums (row 15 of Y).
#pragma unroll
      for (int v = 0; v < 8; ++v)
        ys[16 * col + v + 8 * hi] = acc[v];
      __threadfence_block();
      float o = 0.f, tot = 0.f;
#pragma unroll
      for (int p = 0; p < 16; ++p) {
        const float sp = ys[16 * p + 15];
        if (p < col) o += sp;
        tot += sp;
      }
      // Z = ones(16x1) * o(1x16) + Y : rank-1 offset via one more WMMA.
      v2f a1, b1;
      a1.x = hi ? 0.f : 1.f;  a1.y = 0.f;
      b1.x = hi ? 0.f : o;    b1.y = 0.f;
      acc = __builtin_amdgcn_wmma_f32_16x16x4_f32(
          false, a1, false, b1, (short)0, acc, false, false);
      // Emit: out[b, m, w*8 + ch] = prefix(m) / (m+1)^2
#pragma unroll
      for (int v = 0; v < 8; ++v) {
        const int m = base + 16 * col + v + 8 * hi;
        const float area = (float)((m + 1) * (m + 1));
        out[((size_t)bB * TSIZE + m) * 32 + w * 8 + ch] =
            (acc[v] + carry) / area;
      }
      carry += tot;
      __threadfence_block();           // ys reused next block
    }
  }
}

extern "C" void kernel_launch(void* const* d_in, const int* in_sizes, int n_in,
                              void* d_out, int out_size, void* d_ws, size_t ws_size,
                              hipStream_t stream) {
  (void)in_sizes; (void)n_in; (void)out_size; (void)d_ws; (void)ws_size;
  const float* x = (const float*)d_in[0];
  float* out = (float*)d_out;
  corner_window_mean_kernel<<<dim3(NPLANE), dim3(512), 0, stream>>>(x, out);
}